// AnyonicNeuron_69664369541147
// MI455X (gfx1250) — compile-verified
//
#include <hip/hip_runtime.h>
#include <hip/hip_bf16.h>
#include <math.h>

// ---------------------------------------------------------------------------
// AnyonicNeuron on MI455X (gfx1250, wave32)
//
// reference collapses to:
//   T(2x2 complex) = prod of 28 near-identity braiding matrices (scalar work)
//   braided[0:2]   = T @ wf[0:2]      (maps onto V_WMMA_F32_16X16X4_F32)
//   braided[2:64]  = wf[2:64]         (pure planar->interleaved stream copy)
//   out[c]         = threshold(sigmoid(|diag[c+1] * w[c]|), rand[c])
//
// Bandwidth-bound: ~1.07 GB traffic -> ~46us at 23.3 TB/s. All global traffic
// uses non-temporal hints (no reuse, footprint >> 192MB L2).
// ---------------------------------------------------------------------------

typedef __attribute__((ext_vector_type(2))) float v2f;
typedef __attribute__((ext_vector_type(4))) float v4f;
typedef __attribute__((ext_vector_type(8))) float v8f;

#define EXCH (0.78539816339744830962f) /* pi/4 */

// ---------------------------------------------------------------------------
// Kernel 1: fold the 28 braiding matrices into one 2x2 complex matrix T,
// emit the equivalent real 4x4 matrix Mr (acting on [r0re,r0im,r1re,r1im])
// into ws[0..15], plus raw T10/T11 into ws[16..19] for the readout kernel.
// ---------------------------------------------------------------------------
__global__ void braid_matrix_kernel(const float* __restrict__ angles,
                                    float* __restrict__ ws) {
  if (threadIdx.x != 0 || blockIdx.x != 0) return;
  float t00r = 1.f, t00i = 0.f, t01r = 0.f, t01i = 0.f;
  float t10r = 0.f, t10i = 0.f, t11r = 1.f, t11i = 0.f;
  for (int i = 0; i < 8; ++i) {
    for (int j = i + 1; j < 8; ++j) {
      float ph = angles[i * 8 + j] * EXCH;
      float c = __cosf(ph), s = __sinf(ph);
      float ar = c - 1.f, ai = s;    // e^{i ph} - 1
      float br = c - 1.f, bi = -s;   // conj(e^{i ph}) - 1
      // T <- M * T,  M = [[1, a],[b, 1]]
      float n00r = t00r + ar * t10r - ai * t10i;
      float n00i = t00i + ar * t10i + ai * t10r;
      float n01r = t01r + ar * t11r - ai * t11i;
      float n01i = t01i + ar * t11i + ai * t11r;
      float n10r = br * t00r - bi * t00i + t10r;
      float n10i = br * t00i + bi * t00r + t10i;
      float n11r = br * t01r - bi * t01i + t11r;
      float n11i = br * t01i + bi * t01r + t11i;
      t00r = n00r; t00i = n00i; t01r = n01r; t01i = n01i;
      t10r = n10r; t10i = n10i; t11r = n11r; t11i = n11i;
    }
  }
  // Real 4x4 equivalent: [n0re;n0im;n1re;n1im] = Mr @ [r0re;r0im;r1re;r1im]
  ws[0]  = t00r; ws[1]  = -t00i; ws[2]  = t01r; ws[3]  = -t01i;
  ws[4]  = t00i; ws[5]  =  t00r; ws[6]  = t01i; ws[7]  =  t01r;
  ws[8]  = t10r; ws[9]  = -t10i; ws[10] = t11r; ws[11] = -t11i;
  ws[12] = t10i; ws[13] =  t10r; ws[14] = t11i; ws[15] =  t11r;
  // raw T row 1 for readout (diag element [1,1])
  ws[16] = t10r; ws[17] = t10i; ws[18] = t11r; ws[19] = t11i;
}

// ---------------------------------------------------------------------------
// Kernel 2: rows 0/1 transform via V_WMMA_F32_16X16X4_F32.
// One wave handles one 16-column tile: D(16x16) = A(16x4) x B(4x16).
//   A rows 0..3 = Mr (rest zero)   lane layout: lanes 0-15 K={0,1}, 16-31 K={2,3}
//   B rows (K)  = [r0re,r0im,r1re,r1im] for 16 columns
//   D VGPR v    = output row v on lanes 0-15  -> {d0,d1}=row0 (re,im),
//                                               {d2,d3}=row1 (re,im)
// Stores are interleaved-complex float2 per lane (lanes 0-15).
// ---------------------------------------------------------------------------
__global__ void braid_rows01_wmma_kernel(const float* __restrict__ wf_real,
                                         const float* __restrict__ wf_imag,
                                         const float* __restrict__ ws,
                                         float* __restrict__ braided,
                                         long long n) {
  const int lane = threadIdx.x & 31;
  const long long wave =
      (((long long)blockIdx.x * blockDim.x) + threadIdx.x) >> 5;
  const long long col0 = wave * 16;
  if (col0 >= n) return;  // wave-uniform: EXEC stays all-ones for WMMA

  // A operand (transform matrix), 2 VGPRs
  const int row = lane & 15;
  const int kb = (lane >> 4) << 1;  // K base: 0 or 2
  v2f A;
  A.x = (row < 4) ? ws[row * 4 + kb] : 0.f;
  A.y = (row < 4) ? ws[row * 4 + kb + 1] : 0.f;

  // B operand (wavefunction block), 2 VGPRs
  const int srcRow = lane >> 4;                 // wf row 0 or 1
  const long long col = col0 + (lane & 15);
  const long long base = (long long)srcRow * n + col;
  v2f B;
  B.x = __builtin_nontemporal_load(wf_real + base);   // K=0 / K=2
  B.y = __builtin_nontemporal_load(wf_imag + base);   // K=1 / K=3

  v8f C = {};
  C = __builtin_amdgcn_wmma_f32_16x16x4_f32(
      /*neg_a=*/false, A, /*neg_b=*/false, B,
      /*c_mod=*/(short)0, C, /*reuse_a=*/false, /*reuse_b=*/false);

  if (lane < 16) {
    v2f o0; o0.x = C[0]; o0.y = C[1];   // new row0: re, im
    v2f o1; o1.x = C[2]; o1.y = C[3];   // new row1: re, im
    float* p0 = braided + 2 * col;            // row 0, interleaved complex
    float* p1 = braided + 2 * n + 2 * col;    // row 1
    __builtin_nontemporal_store(o0, (v2f*)p0);
    __builtin_nontemporal_store(o1, (v2f*)p1);
  }
}

// ---------------------------------------------------------------------------
// Kernel 3: rows 2..63 planar->interleaved streaming copy (97% of traffic).
// float4 in from each plane, 2x float4 interleaved out, all non-temporal.
// grid: x over column quads, y over rows (2..63).
// ---------------------------------------------------------------------------
__global__ void copy_interleave_kernel(const float* __restrict__ wf_real,
                                       const float* __restrict__ wf_imag,
                                       float* __restrict__ braided,
                                       long long n) {
  const long long q = (long long)blockIdx.x * blockDim.x + threadIdx.x;
  if (q >= (n >> 2)) return;
  const long long r = 2 + (long long)blockIdx.y;
  const long long c4 = q << 2;
  const long long src = r * n + c4;
  v4f vr = __builtin_nontemporal_load((const v4f*)(wf_real + src));
  v4f vi = __builtin_nontemporal_load((const v4f*)(wf_imag + src));
  v4f o0; o0.x = vr.x; o0.y = vi.x; o0.z = vr.y; o0.w = vi.y;
  v4f o1; o1.x = vr.z; o1.y = vi.z; o1.z = vr.w; o1.w = vi.w;
  float* base = braided + r * 2 * n + 2 * c4;
  __builtin_nontemporal_store(o0, (v4f*)base);
  __builtin_nontemporal_store(o1, (v4f*)(base + 4));
}

// ---------------------------------------------------------------------------
// Kernel 4: readout. out[c] uses diag element [c+1, c+1] of braided:
//   c+1 == 1  -> row-1 transform at column 1 (recomputed from T, exact)
//   c+1 >= 2  -> untouched wf diagonal
//   c   == 63 -> proj = 0
// ---------------------------------------------------------------------------
__global__ void readout_kernel(const float* __restrict__ wf_real,
                               const float* __restrict__ wf_imag,
                               const float* __restrict__ weights,
                               const float* __restrict__ rand_vals,
                               const float* __restrict__ ws,
                               float* __restrict__ out, long long n) {
  const int c = threadIdx.x;
  if (c >= 64) return;
  const int idx = c + 1;
  float dr = 0.f, di = 0.f;
  if (idx == 1) {
    float w0r = wf_real[1],     w0i = wf_imag[1];
    float w1r = wf_real[n + 1], w1i = wf_imag[n + 1];
    float t10r = ws[16], t10i = ws[17], t11r = ws[18], t11i = ws[19];
    dr = t10r * w0r - t10i * w0i + t11r * w1r - t11i * w1i;
    di = t10r * w0i + t10i * w0r + t11r * w1i + t11i * w1r;
  } else if (idx < 64) {
    dr = wf_real[(long long)idx * n + idx];
    di = wf_imag[(long long)idx * n + idx];
  }
  float weighted = sqrtf(dr * dr + di * di) * fabsf(weights[c]);
  float prob = 1.f / (1.f + expf(-weighted));
  out[c] = (rand_vals[c] < prob) ? 0.f : 1.f;  // (R - R_LOW)/(R_HIGH - R_LOW)
}

// ---------------------------------------------------------------------------
extern "C" void kernel_launch(void* const* d_in, const int* in_sizes, int n_in,
                              void* d_out, int out_size, void* d_ws,
                              size_t ws_size, hipStream_t stream) {
  const float* wf_real  = (const float*)d_in[0];
  const float* wf_imag  = (const float*)d_in[1];
  const float* angles   = (const float*)d_in[2];
  const float* weights  = (const float*)d_in[3];
  const float* randv    = (const float*)d_in[4];
  // d_in[5]/d_in[6] (topological_charge / edge_currents) are unused zeros.

  const long long n = (long long)in_sizes[0] / 64;  // wide axis (1048576)
  float* out     = (float*)d_out;       // [64] readout
  float* braided = out + 64;            // [64, n] complex64, interleaved
  float* ws      = (float*)d_ws;        // 20 floats: Mr(16) + T10/T11(4)

  // 1) fold braid chain into 2x2 complex T / real 4x4
  braid_matrix_kernel<<<1, 32, 0, stream>>>(angles, ws);

  // 2) rows 0/1 via WMMA: one wave per 16-column tile
  {
    long long waves = (n + 15) / 16;
    long long threads = waves * 32;
    int block = 256;
    long long grid = (threads + block - 1) / block;
    braid_rows01_wmma_kernel<<<(unsigned)grid, block, 0, stream>>>(
        wf_real, wf_imag, ws, braided, n);
  }

  // 3) rows 2..63 streaming interleave copy
  {
    long long quads = n >> 2;
    int block = 256;
    dim3 grid((unsigned)((quads + block - 1) / block), 62, 1);
    copy_interleave_kernel<<<grid, block, 0, stream>>>(wf_real, wf_imag,
                                                       braided, n);
  }

  // 4) readout
  readout_kernel<<<1, 64, 0, stream>>>(wf_real, wf_imag, weights, randv, ws,
                                       out, n);
}